// TorsionToCartesian_86285892977086
// MI455X (gfx1250) — compile-verified
//
#include <hip/hip_runtime.h>

// ---------------------------------------------------------------------------
// TorsionToCartesian on MI455X (gfx1250, wave32).
//
// Strategy: each residue step is a rigid transform M_i(phi,psi,omega) in the
// canonical frame (every N/CA/C triple is congruent to the seed triple), so
// the chain is the prefix product F_i = M_1 * ... * M_i  -> parallel scan.
// Single 1024-thread workgroup (32 waves, one WGP):
//   Phase A: thread t builds+composes 8 local transforms -> aggregate in LDS.
//   Phase B: Hillis-Steele scan over 1024 aggregates; the combine (4x4 matrix
//            product) is batched 4-at-a-time through V_WMMA_F32_16X16X4_F32
//            using block-diagonal packing (D's diagonal 4x4 blocks = the four
//            independent products).
//   Phase C: thread applies its exclusive prefix, re-walks its 8 steps, emits
//            atoms: N = t, CA = t + 1.46*col0, C = t + cx*col0 + cy*col1.
// All normalizations in the local build are constant multiplies (bond lengths
// and angles fixed); the only transcendentals are 3 sincos per residue, done
// with the hardware v_sin_f32/v_cos_f32 (__sincosf) since torsions are N(0,1)
// radians (no range-reduction needed) and the kernel is latency-bound on this
// arithmetic (memory traffic ~400KB vs 23.3 TB/s HBM).
// ---------------------------------------------------------------------------

typedef float v2f __attribute__((ext_vector_type(2)));
typedef float v8f __attribute__((ext_vector_type(8)));

#define NRES   8192
#define NTHR   1024
#define EPT    8          // residues per thread
#define XSTR   12         // floats per stored 3x4 transform in LDS
#define IDSLOT NTHR       // identity matrix slot

#define CX       2.00852676f   // 1.46 - 1.52*cos(1.94)
#define CYC      1.41757483f   // 1.52*sin(1.94)
#define BC1X     0.36087287f   // (CX-1.46)/1.52
#define BC1Y     0.93261502f   // CYC/1.52
#define M1X     (-0.93261502f)
#define M1Y      0.36087287f
#define D01      0.58950184f   // -1.33*cos(2.03)
#define S1       1.19221963f   //  1.33*sin(2.03)
#define D02      0.77454666f   // -1.46*cos(2.13)
#define S2       1.23760966f   //  1.46*sin(2.13)
#define D03      0.54852676f   // -1.52*cos(1.94)
#define S3       1.41757483f   //  1.52*sin(1.94)
#define INV_L_CN   (1.0f/1.33f)
#define INV_L_NCA  (1.0f/1.46f)
#define INV_SA1    (1.0f/0.89640574f)  // 1/sin(2.03)
#define INV_SA2    (1.0f/0.84767785f)  // 1/sin(2.13)
#define INV_CY     (1.0f/1.41757483f)
#define KNX      (CX + BC1X*D01)
#define KNY      (CYC + BC1Y*D01)

// c = a * b   (3x4 affine, row-major, implicit bottom row [0 0 0 1])
__device__ __forceinline__ void xcompose(const float* __restrict__ a,
                                         const float* __restrict__ b,
                                         float* __restrict__ c) {
#pragma unroll
  for (int r = 0; r < 3; ++r) {
    float a0 = a[r*4+0], a1 = a[r*4+1], a2 = a[r*4+2], a3 = a[r*4+3];
#pragma unroll
    for (int col = 0; col < 3; ++col)
      c[r*4+col] = a0*b[0*4+col] + a1*b[1*4+col] + a2*b[2*4+col];
    c[r*4+3] = a0*b[3] + a1*b[7] + a2*b[11] + a3;
  }
}

__device__ __forceinline__ void set_identity(float* __restrict__ M) {
  M[0]=1.f;M[1]=0.f;M[2]=0.f;M[3]=0.f;
  M[4]=0.f;M[5]=1.f;M[6]=0.f;M[7]=0.f;
  M[8]=0.f;M[9]=0.f;M[10]=1.f;M[11]=0.f;
}

// Local step transform in canonical coordinates (idx==0 -> identity).
__device__ __forceinline__ void build_local(const float* __restrict__ tor,
                                            int idx, float* __restrict__ M) {
  if (idx == 0) { set_identity(M); return; }
  float phi   = tor[idx*3 + 0];
  float psi   = tor[idx*3 + 1];
  float omega = tor[idx*3 - 1];   // torsions[idx-1][2]

  // --- build N' from canonical (N,CA,C): frame vectors are constants -------
  float so, co; __sincosf(omega, &so, &co);   // hardware v_sin/v_cos
  float t1c = S1*co, t1s = S1*so;
  float Nx = KNX + M1X*t1c;
  float Ny = KNY + M1Y*t1c;
  float Nz = t1s;

  // --- build CA' ------------------------------------------------------------
  float b2x = (Nx - CX)  * INV_L_CN;   // unit C->N'
  float b2y = (Ny - CYC) * INV_L_CN;
  float b2z =  Nz        * INV_L_CN;
  float n2x = ( BC1Y * b2z) * INV_SA1;             // cross(BC1,b2)/sin
  float n2y = (-BC1X * b2z) * INV_SA1;
  float n2z = ( BC1X * b2y - BC1Y * b2x) * INV_SA1;
  float m2x = n2y*b2z - n2z*b2y;
  float m2y = n2z*b2x - n2x*b2z;
  float m2z = n2x*b2y - n2y*b2x;
  float sp, cp; __sincosf(phi, &sp, &cp);
  float t2c = S2*cp, t2s = S2*sp;
  float Ax = Nx + b2x*D02 + m2x*t2c + n2x*t2s;
  float Ay = Ny + b2y*D02 + m2y*t2c + n2y*t2s;
  float Az = Nz + b2z*D02 + m2z*t2c + n2z*t2s;

  // --- build C' -------------------------------------------------------------
  float b3x = (Ax - Nx) * INV_L_NCA;   // unit N'->CA'  (== frame e1)
  float b3y = (Ay - Ny) * INV_L_NCA;
  float b3z = (Az - Nz) * INV_L_NCA;
  float n3x = (b2y*b3z - b2z*b3y) * INV_SA2;
  float n3y = (b2z*b3x - b2x*b3z) * INV_SA2;
  float n3z = (b2x*b3y - b2y*b3x) * INV_SA2;
  float m3x = n3y*b3z - n3z*b3y;
  float m3y = n3z*b3x - n3x*b3z;
  float m3z = n3x*b3y - n3y*b3x;
  float ss, cs; __sincosf(psi, &ss, &cs);
  float t3c = S3*cs, t3s = S3*ss;
  float Cx = Ax + b3x*D03 + m3x*t3c + n3x*t3s;
  float Cy = Ay + b3y*D03 + m3y*t3c + n3y*t3s;
  float Cz = Az + b3z*D03 + m3z*t3c + n3z*t3s;

  // --- orthonormal frame of the new triple (e1 = b3, |perp| = CYC exactly) --
  float ux = Cx - Nx, uy = Cy - Ny, uz = Cz - Nz;
  float du = ux*b3x + uy*b3y + uz*b3z;
  float e2x = (ux - du*b3x) * INV_CY;
  float e2y = (uy - du*b3y) * INV_CY;
  float e2z = (uz - du*b3z) * INV_CY;
  float e3x = b3y*e2z - b3z*e2y;
  float e3y = b3z*e2x - b3x*e2z;
  float e3z = b3x*e2y - b3y*e2x;

  M[0]=b3x; M[1]=e2x; M[2]=e3x;  M[3]=Nx;
  M[4]=b3y; M[5]=e2y; M[6]=e3y;  M[7]=Ny;
  M[8]=b3z; M[9]=e2z; M[10]=e3z; M[11]=Nz;
}

__global__ __launch_bounds__(NTHR)
void torsion_nerf_kernel(const float* __restrict__ tor, float* __restrict__ out) {
  __shared__ float smem[(NTHR + 1) * XSTR];   // 49.2 KB (of 320 KB WGP LDS)
  const int tid = threadIdx.x;

  // ---------------- Phase A: per-thread serial aggregate -------------------
  {
    float S[12], Mx[12], T[12];
    build_local(tor, tid*EPT, S);
    for (int j = 1; j < EPT; ++j) {
      build_local(tor, tid*EPT + j, Mx);
      xcompose(S, Mx, T);
#pragma unroll
      for (int i = 0; i < 12; ++i) S[i] = T[i];
    }
#pragma unroll
    for (int i = 0; i < 12; ++i) smem[tid*XSTR + i] = S[i];
    if (tid == 0) {
      float I[12]; set_identity(I);
#pragma unroll
      for (int i = 0; i < 12; ++i) smem[IDSLOT*XSTR + i] = I[i];
    }
  }
  __syncthreads();

  // ---------------- Phase B: Hillis-Steele scan via WMMA -------------------
  // Combine T[i] <- T[i-d] * T[i].  Four 4x4 products per
  // v_wmma_f32_16x16x4_f32 via block-diagonal packing:
  //   A(16x4)  = [L0;L1;L2;L3]  (stacked left matrices)
  //   B(4x16)  = [R0 R1 R2 R3]  (concatenated right matrices)
  //   D(16x16) block(k,k) = Lk*Rk  (diagonal blocks extracted).
  // A layout (f32 16x4): lane l<16 holds row (l), cols {0,1}; lane l>=16 holds
  // row (l-16), cols {2,3}.  B layout (4x16, mirroring C/D striping): VGPR0 =
  // rows {0 | 2}, VGPR1 = rows {1 | 3} across the two lane halves.
  // C/D layout: VGPR r, lanes 0-15 -> (M=r, N=lane); lanes 16-31 -> (M=r+8).
  {
    const int l    = tid & 31;
    const int base = (tid >> 5) * 32;       // this wave's 32 elements
    const int kk   = (l & 15) >> 2;         // block index within group
    const int jj   = l & 3;                 // row (A) / col (B) within block
    const bool hi  = (l >= 16);
    const int c0   = hi ? 2 : 0;            // A cols / B rows start
    // diagonal extraction mapping
    const bool act  = (l < 8) || (l >= 24);
    const int  kd   = (l < 8) ? (l >> 2) : (2 + ((l >> 2) - 6));
    const int  nd   = l & 3;
    const bool rsel = (l & 4) != 0;

    for (int d = 1; d < NTHR; d <<= 1) {
      v2f aOp[8], bOp[8];
#pragma unroll
      for (int g = 0; g < 8; ++g) {
        int eA = base + 4*g + kk - d;
        if (eA < 0) eA = IDSLOT;            // compose with identity
        v2f a;
        if (jj == 3) {                      // implicit bottom row [0 0 0 1]
          a.x = 0.f; a.y = hi ? 1.f : 0.f;
        } else {
          const float* p = &smem[eA*XSTR + jj*4 + c0];
          a.x = p[0]; a.y = p[1];
        }
        aOp[g] = a;
        int eB = base + 4*g + kk;
        v2f b;
        b.x = smem[eB*XSTR + c0*4 + jj];    // rows 0 / 2 (always stored)
        b.y = hi ? ((jj == 3) ? 1.f : 0.f)  // row 3 implicit
                 : smem[eB*XSTR + 4 + jj];  // row 1
        bOp[g] = b;
      }
      __syncthreads();                      // loads done before anyone stores
#pragma unroll
      for (int g = 0; g < 8; ++g) {
        v8f cz = {};
        v8f dd = __builtin_amdgcn_wmma_f32_16x16x4_f32(
            false, aOp[g], false, bOp[g], (short)0, cz, false, false);
        if (act) {
          int eD = base + 4*g + kd;
          float r0 = rsel ? dd[4] : dd[0];  // result rows 0..2 of block kd,
          float r1 = rsel ? dd[5] : dd[1];  // column nd (row 3 implicit)
          float r2 = rsel ? dd[6] : dd[2];
          smem[eD*XSTR + 0 + nd] = r0;
          smem[eD*XSTR + 4 + nd] = r1;
          smem[eD*XSTR + 8 + nd] = r2;
        }
      }
      __syncthreads();
    }
  }

  // ---------------- Phase C: apply exclusive prefix, emit atoms ------------
  {
    float G[12], Mx[12], T[12];
    if (tid == 0) set_identity(G);
    else {
#pragma unroll
      for (int i = 0; i < 12; ++i) G[i] = smem[(tid-1)*XSTR + i];
    }
    for (int j = 0; j < EPT; ++j) {
      int idx = tid*EPT + j;
      build_local(tor, idx, Mx);
      xcompose(G, Mx, T);
#pragma unroll
      for (int i = 0; i < 12; ++i) G[i] = T[i];
      float* o = out + idx*9;
      float nx = G[3], ny = G[7], nz = G[11];
      o[0] = nx;                         o[1] = ny;                         o[2] = nz;
      o[3] = nx + 1.46f*G[0];            o[4] = ny + 1.46f*G[4];            o[5] = nz + 1.46f*G[8];
      o[6] = nx + CX*G[0] + CYC*G[1];    o[7] = ny + CX*G[4] + CYC*G[5];    o[8] = nz + CX*G[8] + CYC*G[9];
    }
  }
}

extern "C" void kernel_launch(void* const* d_in, const int* in_sizes, int n_in,
                              void* d_out, int out_size, void* d_ws, size_t ws_size,
                              hipStream_t stream) {
  (void)in_sizes; (void)n_in; (void)d_ws; (void)ws_size; (void)out_size;
  const float* tor = (const float*)d_in[0];
  float* out = (float*)d_out;
  torsion_nerf_kernel<<<dim3(1), dim3(NTHR), 0, stream>>>(tor, out);
}